// RelGATv2EncoderTG_54047868453345
// MI455X (gfx1250) — compile-verified
//
#include <hip/hip_runtime.h>
#include <math.h>

#define NN    32768
#define EE    262144
#define DDIM  128
#define HHEAD 4
#define CCH   128
#define HC    512
#define RDIM  64
#define NRELS 64
#define NLAY  2
#define LEAKS 0.2f
#define LNEPS 1e-5f

typedef float v2f __attribute__((ext_vector_type(2)));
typedef float v8f __attribute__((ext_vector_type(8)));

// ---- monotonic int encoding for float atomic-max ----
__device__ __forceinline__ int fenc(float f) {
  int i = __float_as_int(f);
  return i >= 0 ? i : (int)(i ^ 0x7fffffff);
}
__device__ __forceinline__ float fdec(int e) {
  int b = e >= 0 ? e : (int)(e ^ 0x7fffffff);
  return __int_as_float(b);
}
#define ENC_NEG_INF ((int)0x807fffff)  // fenc(-inf)

// =====================================================================
// f32 WMMA GEMM: Dst[M,Ncols] = A[M,K] * B[K,Ncols] (+ bias[Ncols])
// One wave computes a 16x64 tile (4 x V_WMMA_F32_16X16X4_F32 accumulators).
// M % 16 == 0, K % 4 == 0, Ncols % 64 == 0.
// =====================================================================
__global__ void gemm_f32_wmma(const float* __restrict__ A,
                              const float* __restrict__ B,
                              const float* __restrict__ bias,
                              float* __restrict__ Dst,
                              int M, int K, int Ncols) {
  const int lane = threadIdx.x & 31;
  const int wave = threadIdx.x >> 5;
  const int ngroups = Ncols >> 6;          // 64-col groups
  const int mtiles  = M >> 4;
  int w = blockIdx.x * 8 + wave;           // 8 waves / block
  int mtile = w / ngroups;
  int ng    = w % ngroups;
  if (mtile >= mtiles) return;             // wave-uniform exit (EXEC all-1 inside)

  const int m0   = mtile << 4;
  const int n0   = ng << 6;
  const int half = lane >> 4;              // 0: lanes 0-15, 1: lanes 16-31
  const int lm   = lane & 15;

  v8f acc0 = {}, acc1 = {}, acc2 = {}, acc3 = {};
  const float* arow = A + (size_t)(m0 + lm) * K;

  for (int k0 = 0; k0 < K; k0 += 4) {
    const int ka = k0 + 2 * half;
    // A fragment: 16x4 f32; lane half selects K pair {ka, ka+1}
    v2f a;
    a.x = arow[ka];
    a.y = arow[ka + 1];
    // B fragments: 4x16 f32 each; row ka in .x, row ka+1 in .y
    const float* bp = B + (size_t)ka * Ncols + n0 + lm;
    v2f b0, b1, b2, b3;
    b0.x = bp[0];   b0.y = bp[Ncols + 0];
    b1.x = bp[16];  b1.y = bp[Ncols + 16];
    b2.x = bp[32];  b2.y = bp[Ncols + 32];
    b3.x = bp[48];  b3.y = bp[Ncols + 48];
    acc0 = __builtin_amdgcn_wmma_f32_16x16x4_f32(false, a, false, b0, (short)0, acc0, false, false);
    acc1 = __builtin_amdgcn_wmma_f32_16x16x4_f32(false, a, false, b1, (short)0, acc1, false, false);
    acc2 = __builtin_amdgcn_wmma_f32_16x16x4_f32(false, a, false, b2, (short)0, acc2, false, false);
    acc3 = __builtin_amdgcn_wmma_f32_16x16x4_f32(false, a, false, b3, (short)0, acc3, false, false);
  }

  float bc0 = 0.f, bc1 = 0.f, bc2 = 0.f, bc3 = 0.f;
  if (bias) {
    bc0 = bias[n0 + lm];
    bc1 = bias[n0 + 16 + lm];
    bc2 = bias[n0 + 32 + lm];
    bc3 = bias[n0 + 48 + lm];
  }
  // D layout: VGPR r -> rows m0+r (lanes 0-15) / m0+8+r (lanes 16-31), col = n0+t*16+lm
#pragma unroll
  for (int r = 0; r < 8; ++r) {
    float* dr = Dst + (size_t)(m0 + half * 8 + r) * Ncols + n0 + lm;
    dr[0]  = acc0[r] + bc0;
    dr[16] = acc1[r] + bc1;
    dr[32] = acc2[r] + bc2;
    dr[48] = acc3[r] + bc3;
  }
}

// =====================================================================
// Utility kernels
// =====================================================================
__global__ void zero_f32_k(float* __restrict__ p, long n) {
  long i = (long)blockIdx.x * blockDim.x + threadIdx.x;
  if (i < n) p[i] = 0.f;
}

__global__ void init_h_k(const float* __restrict__ x, const unsigned char* __restrict__ mask,
                         float* __restrict__ h) {
  long i = (long)blockIdx.x * blockDim.x + threadIdx.x;
  if (i >= (long)NN * DDIM) return;
  int node = (int)(i >> 7);
  h[i] = mask[node] ? x[i] : 0.f;
}

__global__ void write_out_k(const float* __restrict__ h, const unsigned char* __restrict__ mask,
                            float* __restrict__ out) {
  long i = (long)blockIdx.x * blockDim.x + threadIdx.x;
  if (i >= (long)NN * DDIM) return;
  int node = (int)(i >> 7);
  out[i] = mask[node] ? h[i] : 0.f;
}

// per-(dst,rel) counts + in-degree (replaces 64 atomic adds/edge with 2)
__global__ void count_edges_k(const int* __restrict__ ei, const int* __restrict__ ety,
                              float* __restrict__ cnt, float* __restrict__ deg) {
  int e = blockIdx.x * blockDim.x + threadIdx.x;
  if (e >= EE) return;
  int d = ei[EE + e];
  int t = ety[e];
  atomicAdd(&cnt[(size_t)d * NRELS + t], 1.f);
  atomicAdd(&deg[d], 1.f);
}

// sattr[i,r] /= max(deg[i],1)   (sattr holds cnt @ rel_emb from WMMA GEMM)
__global__ void scale_sattr_k(float* __restrict__ sattr, const float* __restrict__ deg) {
  long i = (long)blockIdx.x * blockDim.x + threadIdx.x;
  if (i >= (long)NN * RDIM) return;
  int node = (int)(i / RDIM);
  sattr[i] *= 1.f / fmaxf(deg[node], 1.f);
}

__global__ void init_md_k(int* __restrict__ mmax, float* __restrict__ denom) {
  int i = blockIdx.x * blockDim.x + threadIdx.x;
  if (i >= NN * HHEAD) return;
  mmax[i] = ENC_NEG_INF;
  denom[i] = 0.f;
}

// =====================================================================
// Pass 1: logits[e,h] = sum_c att[h,c]*leaky(xl[src]+xr[dst]+ep)
// one wave per edge entry; lanes cover 16 contiguous channels each
// (8 lanes per head -> shfl reduction of width 8). Entries E..E+N-1 are
// self loops (ep = self_ep[node]).
// =====================================================================
__global__ void edge_logits_k(const float* __restrict__ xl, const float* __restrict__ xr,
                              const float* __restrict__ sep, const float* __restrict__ relproj,
                              const float* __restrict__ att,
                              const int* __restrict__ ei, const int* __restrict__ ety,
                              float* __restrict__ logits, int* __restrict__ mmax) {
  long gt = (long)blockIdx.x * blockDim.x + threadIdx.x;
  int widx = (int)(gt >> 5);
  int lane = (int)(gt & 31);
  if (widx >= EE + NN) return;

  int s, d;
  const float* ep;
  if (widx < EE) {
    s = ei[widx];
    d = ei[EE + widx];
    ep = relproj + (size_t)ety[widx] * HC;
  } else {
    s = d = widx - EE;
    ep = sep + (size_t)s * HC;
  }
  const float* pl = xl + (size_t)s * HC;
  const float* pr = xr + (size_t)d * HC;

  const int j0 = lane << 4;       // 16 channels per lane
  const int hh = j0 >> 7;         // head index
  float partial = 0.f;
#pragma unroll
  for (int q = 0; q < 4; ++q) {
    int j = j0 + q * 4;
    float4 a = *(const float4*)(pl + j);
    float4 b = *(const float4*)(pr + j);
    float4 c = *(const float4*)(ep + j);
    float4 wv = *(const float4*)(att + j);
    float e0 = a.x + b.x + c.x; e0 = e0 > 0.f ? e0 : LEAKS * e0;
    float e1 = a.y + b.y + c.y; e1 = e1 > 0.f ? e1 : LEAKS * e1;
    float e2 = a.z + b.z + c.z; e2 = e2 > 0.f ? e2 : LEAKS * e2;
    float e3 = a.w + b.w + c.w; e3 = e3 > 0.f ? e3 : LEAKS * e3;
    partial += wv.x * e0 + wv.y * e1 + wv.z * e2 + wv.w * e3;
  }
  // reduce within 8-lane head group
  partial += __shfl_down(partial, 4, 8);
  partial += __shfl_down(partial, 2, 8);
  partial += __shfl_down(partial, 1, 8);
  if ((lane & 7) == 0) {
    logits[(size_t)widx * HHEAD + hh] = partial;
    atomicMax(&mmax[(size_t)d * HHEAD + hh], fenc(partial));
  }
}

// =====================================================================
// Pass 2: p = exp(logit - m[dst]); denom[dst,h] += p;
//         accum[dst,:] += p * xl[src,:]
// =====================================================================
__global__ void edge_accum_k(const float* __restrict__ xl,
                             const int* __restrict__ ei,
                             const float* __restrict__ logits, const int* __restrict__ mmax,
                             float* __restrict__ denom, float* __restrict__ accum) {
  long gt = (long)blockIdx.x * blockDim.x + threadIdx.x;
  int widx = (int)(gt >> 5);
  int lane = (int)(gt & 31);
  if (widx >= EE + NN) return;

  int s, d;
  if (widx < EE) { s = ei[widx]; d = ei[EE + widx]; }
  else           { s = d = widx - EE; }

  const int j0 = lane << 4;
  const int hh = j0 >> 7;
  float m  = fdec(mmax[(size_t)d * HHEAD + hh]);
  float lg = logits[(size_t)widx * HHEAD + hh];
  float p  = expf(lg - m);
  if ((lane & 7) == 0) atomicAdd(&denom[(size_t)d * HHEAD + hh], p);

  const float* pl = xl + (size_t)s * HC + j0;
  float* pa = accum + (size_t)d * HC + j0;
#pragma unroll
  for (int q = 0; q < 16; ++q) atomicAdd(&pa[q], p * pl[q]);
}

// =====================================================================
// Per-node epilogue: head-mean(+bias) -> GELU -> residual -> LayerNorm
// one wave per node; lane handles 4 channels; butterfly reductions.
// =====================================================================
__global__ void node_update_k(float* __restrict__ h,
                              const float* __restrict__ accum, const float* __restrict__ denom,
                              const float* __restrict__ bias_o,
                              const float* __restrict__ ln_g, const float* __restrict__ ln_b) {
  long gt = (long)blockIdx.x * blockDim.x + threadIdx.x;
  int i = (int)(gt >> 5);
  int lane = (int)(gt & 31);
  if (i >= NN) return;

  float dinv[HHEAD];
#pragma unroll
  for (int hh = 0; hh < HHEAD; ++hh)
    dinv[hh] = 1.f / (denom[(size_t)i * HHEAD + hh] + 1e-16f);

  float vals[4];
  float sum = 0.f, sumsq = 0.f;
#pragma unroll
  for (int q = 0; q < 4; ++q) {
    int dch = lane * 4 + q;
    float s = 0.f;
#pragma unroll
    for (int hh = 0; hh < HHEAD; ++hh)
      s += accum[(size_t)i * HC + hh * CCH + dch] * dinv[hh];
    s = s * 0.25f + bias_o[dch];
    float g = 0.5f * s * (1.f + erff(s * 0.70710678118654752f));  // exact GELU
    float hn = h[(size_t)i * DDIM + dch] + g;
    vals[q] = hn;
    sum += hn;
    sumsq += hn * hn;
  }
#pragma unroll
  for (int off = 16; off > 0; off >>= 1) {
    sum   += __shfl_xor(sum, off, 32);
    sumsq += __shfl_xor(sumsq, off, 32);
  }
  float mu  = sum * (1.f / DDIM);
  float var = sumsq * (1.f / DDIM) - mu * mu;
  float rstd = rsqrtf(var + LNEPS);
#pragma unroll
  for (int q = 0; q < 4; ++q) {
    int dch = lane * 4 + q;
    h[(size_t)i * DDIM + dch] = (vals[q] - mu) * rstd * ln_g[dch] + ln_b[dch];
  }
}

// =====================================================================
// Host orchestration
// =====================================================================
static inline int cdiv(long a, long b) { return (int)((a + b - 1) / b); }

extern "C" void kernel_launch(void* const* d_in, const int* in_sizes, int n_in,
                              void* d_out, int out_size, void* d_ws, size_t ws_size,
                              hipStream_t stream) {
  const float* x_flat  = (const float*)d_in[0];
  const int*   ei      = (const int*)d_in[1];          // [2,E]
  const int*   ety     = (const int*)d_in[2];          // [E]
  const unsigned char* mask = (const unsigned char*)d_in[3];
  const float* rel_emb = (const float*)d_in[4];        // [NREL,R]
  const float* Wl      = (const float*)d_in[5];        // [L,D,HC]
  const float* bl      = (const float*)d_in[6];
  const float* Wr      = (const float*)d_in[7];
  const float* br      = (const float*)d_in[8];
  const float* We      = (const float*)d_in[9];        // [L,R,HC]
  const float* att     = (const float*)d_in[10];       // [L,H,C]
  const float* bo      = (const float*)d_in[11];       // [L,D]
  const float* lng     = (const float*)d_in[12];
  const float* lnb     = (const float*)d_in[13];
  float* out = (float*)d_out;

  // workspace carve-up
  char* w = (char*)d_ws;
  float* h       = (float*)w; w += (size_t)NN * DDIM * 4;
  float* xl      = (float*)w; w += (size_t)NN * HC * 4;
  float* xr      = (float*)w; w += (size_t)NN * HC * 4;
  float* sep     = (float*)w; w += (size_t)NN * HC * 4;
  float* accum   = (float*)w; w += (size_t)NN * HC * 4;
  float* cnt     = (float*)w; w += (size_t)NN * NRELS * 4;
  float* sattr   = (float*)w; w += (size_t)NN * RDIM * 4;
  float* relproj = (float*)w; w += (size_t)NRELS * HC * 4;
  float* deg     = (float*)w; w += (size_t)NN * 4;
  float* denom   = (float*)w; w += (size_t)NN * HHEAD * 4;
  int*   mmax    = (int*)  w; w += (size_t)NN * HHEAD * 4;
  float* logits  = (float*)w; w += (size_t)(EE + NN) * HHEAD * 4;

  const int TB = 256;
  const long ND = (long)NN * DDIM;
  const long T  = (long)(EE + NN);

  // h = mask ? x : 0
  init_h_k<<<cdiv(ND, TB), TB, 0, stream>>>(x_flat, mask, h);

  // per-(dst,rel) counts + degree
  zero_f32_k<<<cdiv((long)NN * NRELS, TB), TB, 0, stream>>>(cnt, (long)NN * NRELS);
  zero_f32_k<<<cdiv((long)NN, TB), TB, 0, stream>>>(deg, (long)NN);
  count_edges_k<<<cdiv((long)EE, TB), TB, 0, stream>>>(ei, ety, cnt, deg);

  // self-loop edge attr: sattr = (cnt @ rel_emb) / max(deg,1)   [WMMA GEMM]
  gemm_f32_wmma<<<cdiv((long)(NN / 16) * (RDIM / 64), 8), TB, 0, stream>>>(
      cnt, rel_emb, nullptr, sattr, NN, NRELS, RDIM);
  scale_sattr_k<<<cdiv((long)NN * RDIM, TB), TB, 0, stream>>>(sattr, deg);

  for (int l = 0; l < NLAY; ++l) {
    const float* Wl_l = Wl + (size_t)l * DDIM * HC;
    const float* bl_l = bl + (size_t)l * HC;
    const float* Wr_l = Wr + (size_t)l * DDIM * HC;
    const float* br_l = br + (size_t)l * HC;
    const float* We_l = We + (size_t)l * RDIM * HC;
    const float* at_l = att + (size_t)l * HHEAD * CCH;
    const float* bo_l = bo + (size_t)l * DDIM;
    const float* lg_l = lng + (size_t)l * DDIM;
    const float* lb_l = lnb + (size_t)l * DDIM;

    // WMMA GEMMs
    gemm_f32_wmma<<<cdiv((long)(NN / 16) * (HC / 64), 8), TB, 0, stream>>>(
        h, Wl_l, bl_l, xl, NN, DDIM, HC);
    gemm_f32_wmma<<<cdiv((long)(NN / 16) * (HC / 64), 8), TB, 0, stream>>>(
        h, Wr_l, br_l, xr, NN, DDIM, HC);
    gemm_f32_wmma<<<cdiv((long)(NRELS / 16) * (HC / 64), 8), TB, 0, stream>>>(
        rel_emb, We_l, nullptr, relproj, NRELS, RDIM, HC);
    gemm_f32_wmma<<<cdiv((long)(NN / 16) * (HC / 64), 8), TB, 0, stream>>>(
        sattr, We_l, nullptr, sep, NN, RDIM, HC);

    // segment-softmax state
    zero_f32_k<<<cdiv((long)NN * HC, TB), TB, 0, stream>>>(accum, (long)NN * HC);
    init_md_k<<<cdiv((long)NN * HHEAD, TB), TB, 0, stream>>>(mmax, denom);

    // edge passes (one wave / edge entry)
    edge_logits_k<<<cdiv(T * 32, TB), TB, 0, stream>>>(
        xl, xr, sep, relproj, at_l, ei, ety, logits, mmax);
    edge_accum_k<<<cdiv(T * 32, TB), TB, 0, stream>>>(
        xl, ei, logits, mmax, denom, accum);

    // node epilogue (one wave / node)
    node_update_k<<<cdiv((long)NN * 32, TB), TB, 0, stream>>>(
        h, accum, denom, bo_l, lg_l, lb_l);
  }

  write_out_k<<<cdiv(ND, TB), TB, 0, stream>>>(h, mask, out);
}